// GATLayer_25649544692288
// MI455X (gfx1250) — compile-verified
//
#include <hip/hip_runtime.h>
#include <math.h>

typedef __attribute__((ext_vector_type(2))) float v2f;
typedef __attribute__((ext_vector_type(8))) float v8f;

#define IN_DIM    128
#define OUT_DIM   64
#define EDGE_DIM  16
#define NEG_SLOPE 0.2f

__device__ __forceinline__ float lrelu(float x) { return x > 0.0f ? x : NEG_SLOPE * x; }

// sign-aware float atomic max via int/uint ordering (location must be init'd to -inf bits)
__device__ __forceinline__ void atomicMaxFloat(float* addr, float val) {
  if (!(val < 0.0f)) {
    atomicMax((int*)addr, __float_as_int(val));
  } else {
    atomicMin((unsigned int*)addr, __float_as_uint(val));
  }
}

// ---------------------------------------------------------------- init
__global__ void init_kernel(float* __restrict__ out, int out_elems,
                            float* __restrict__ e_max, float* __restrict__ denom,
                            int n_nodes) {
  int i = blockIdx.x * blockDim.x + threadIdx.x;
  if (i < out_elems) out[i] = 0.0f;
  if (i < n_nodes) { e_max[i] = -INFINITY; denom[i] = 0.0f; }
}

// ------------------------------------------------ z = lrelu(h @ fc_w^T), WMMA f32
// One wave per 16x16 output tile; 4 waves per block cover N=64.
// A(16x4 f32): lane holds row M=lane%16, K = 2*(lane/16)+{0,1}  -> contiguous 8B load
// B(4x16 f32): lane holds col N=lane%16, K = 2*(lane/16)+{0,1}; B[k][n]=fc_w[n][k]
// C(16x16 f32): acc[r] -> M = 16*mtile + r + 8*(lane/16), N = 16*ntile + lane%16
__global__ void gemm_wmma(const float* __restrict__ h, const float* __restrict__ fc_w,
                          float* __restrict__ z) {
  const int mtile = blockIdx.x;
  const int ntile = threadIdx.x >> 5;   // 0..3
  const int lane  = threadIdx.x & 31;
  const int half  = lane >> 4;
  const int lrow  = lane & 15;

  const float* arow = h    + (size_t)(mtile * 16 + lrow) * IN_DIM + 2 * half;
  const float* brow = fc_w + (size_t)(ntile * 16 + lrow) * IN_DIM + 2 * half;

  v8f acc = {};
#pragma unroll
  for (int k = 0; k < IN_DIM; k += 4) {
    v2f a = *(const v2f*)(arow + k);
    v2f b = *(const v2f*)(brow + k);
    acc = __builtin_amdgcn_wmma_f32_16x16x4_f32(false, a, false, b,
                                                (short)0, acc, false, false);
  }

  float* zo = z + (size_t)(mtile * 16 + 8 * half) * OUT_DIM + ntile * 16 + lrow;
#pragma unroll
  for (int r = 0; r < 8; ++r)
    zo[(size_t)r * OUT_DIM] = lrelu(acc[r]);
}

// scalar fallback for M-tail (unused for N_NODES=100000, kept for generality)
__global__ void gemm_tail(const float* __restrict__ h, const float* __restrict__ fc_w,
                          float* __restrict__ z, int m0, int n_nodes) {
  int idx = blockIdx.x * blockDim.x + threadIdx.x;
  int m = m0 + idx / OUT_DIM;
  int n = idx % OUT_DIM;
  if (m >= n_nodes) return;
  float s = 0.0f;
  for (int k = 0; k < IN_DIM; ++k) s += h[(size_t)m * IN_DIM + k] * fc_w[(size_t)n * IN_DIM + k];
  z[(size_t)m * OUT_DIM + n] = lrelu(s);
}

// ---------------------------------------------- per-node attention scalars
// s_src[n] = z[n]·a_w[0:64], s_dst[n] = z[n]·a_w[80:144]
__global__ void node_scalars(const float* __restrict__ z, const float* __restrict__ a_w,
                             float* __restrict__ s_src, float* __restrict__ s_dst, int n_nodes) {
  int i = blockIdx.x * blockDim.x + threadIdx.x;
  if (i >= n_nodes) return;
  const float4* zr = (const float4*)(z + (size_t)i * OUT_DIM);
  const float4* as = (const float4*)(a_w);                       // a_src
  const float4* ad = (const float4*)(a_w + OUT_DIM + EDGE_DIM);  // a_dst (offset 80, 16B aligned)
  float ss = 0.0f, sd = 0.0f;
#pragma unroll
  for (int j = 0; j < OUT_DIM / 4; ++j) {
    float4 zv = zr[j], av = as[j], dv = ad[j];
    ss += zv.x * av.x + zv.y * av.y + zv.z * av.z + zv.w * av.w;
    sd += zv.x * dv.x + zv.y * dv.y + zv.z * dv.z + zv.w * dv.w;
  }
  s_src[i] = ss;
  s_dst[i] = sd;
}

// ---------------------------------------------- edge pass 1: e + segment max
__global__ void edge_score(const float* __restrict__ w, const float* __restrict__ a_w,
                           const int* __restrict__ src, const int* __restrict__ dst,
                           const float* __restrict__ s_src, const float* __restrict__ s_dst,
                           float* __restrict__ e, float* __restrict__ e_max, int n_edges) {
  int i = blockIdx.x * blockDim.x + threadIdx.x;
  if (i >= n_edges) return;
  const float4* wr = (const float4*)(w + (size_t)i * EDGE_DIM);
  const float4* ae = (const float4*)(a_w + OUT_DIM);   // a_edge (offset 64, 16B aligned)
  float dotv = 0.0f;
#pragma unroll
  for (int j = 0; j < EDGE_DIM / 4; ++j) {
    float4 wv = wr[j], av = ae[j];
    dotv += wv.x * av.x + wv.y * av.y + wv.z * av.z + wv.w * av.w;
  }
  int d = dst[i];
  float ev = lrelu(s_src[src[i]] + dotv + s_dst[d]);
  e[i] = ev;
  atomicMaxFloat(&e_max[d], ev);
}

// ---------------------------------------------- edge pass 2: ex + segment sum
__global__ void edge_exp(float* __restrict__ e, const float* __restrict__ e_max,
                         float* __restrict__ denom, const int* __restrict__ dst, int n_edges) {
  int i = blockIdx.x * blockDim.x + threadIdx.x;
  if (i >= n_edges) return;
  int d = dst[i];
  float ex = expf(e[i] - e_max[d]);
  e[i] = ex;                // overwrite e with ex
  atomicAdd(&denom[d], ex);
}

// ---------------------------------------------- edge pass 3: weighted scatter
// 16 threads per edge, 4 features each (float4 gather of z[src], 4 f32 atomics to out[dst])
__global__ void edge_scatter(const float* __restrict__ ex, const float* __restrict__ denom,
                             const int* __restrict__ src, const int* __restrict__ dst,
                             const float* __restrict__ z, float* __restrict__ out,
                             long long total) {
  long long t = (long long)blockIdx.x * blockDim.x + threadIdx.x;
  if (t >= total) return;
  int i = (int)(t >> 4);
  int f = (int)(t & 15) * 4;
  int d = dst[i];
  float alpha = ex[i] / denom[d];
  float4 zv = *(const float4*)(z + (size_t)src[i] * OUT_DIM + f);
  float* o = out + (size_t)d * OUT_DIM + f;
  atomicAdd(o + 0, alpha * zv.x);
  atomicAdd(o + 1, alpha * zv.y);
  atomicAdd(o + 2, alpha * zv.z);
  atomicAdd(o + 3, alpha * zv.w);
}

extern "C" void kernel_launch(void* const* d_in, const int* in_sizes, int n_in,
                              void* d_out, int out_size, void* d_ws, size_t ws_size,
                              hipStream_t stream) {
  const float* h    = (const float*)d_in[0];
  const float* w    = (const float*)d_in[1];
  const float* fc_w = (const float*)d_in[2];
  const float* a_w  = (const float*)d_in[3];
  const int*   src  = (const int*)d_in[4];
  const int*   dst  = (const int*)d_in[5];
  float* out = (float*)d_out;

  const int n_nodes = in_sizes[0] / IN_DIM;
  const int n_edges = in_sizes[4];

  // workspace layout (floats): z | s_src | s_dst | e_max | denom | e/ex
  float* ws    = (float*)d_ws;
  float* z     = ws;
  float* s_src = z + (size_t)n_nodes * OUT_DIM;
  float* s_dst = s_src + n_nodes;
  float* e_max = s_dst + n_nodes;
  float* denom = e_max + n_nodes;
  float* ebuf  = denom + n_nodes;

  int init_n = out_size > n_nodes ? out_size : n_nodes;
  init_kernel<<<(init_n + 255) / 256, 256, 0, stream>>>(out, out_size, e_max, denom, n_nodes);

  int mtiles = n_nodes / 16;
  if (mtiles > 0)
    gemm_wmma<<<mtiles, 128, 0, stream>>>(h, fc_w, z);
  int rem = n_nodes - mtiles * 16;
  if (rem > 0)
    gemm_tail<<<(rem * OUT_DIM + 255) / 256, 256, 0, stream>>>(h, fc_w, z, mtiles * 16, n_nodes);

  node_scalars<<<(n_nodes + 255) / 256, 256, 0, stream>>>(z, a_w, s_src, s_dst, n_nodes);

  edge_score<<<(n_edges + 255) / 256, 256, 0, stream>>>(w, a_w, src, dst, s_src, s_dst,
                                                        ebuf, e_max, n_edges);
  edge_exp<<<(n_edges + 255) / 256, 256, 0, stream>>>(ebuf, e_max, denom, dst, n_edges);

  long long total = (long long)n_edges * 16;
  int blocks = (int)((total + 255) / 256);
  edge_scatter<<<blocks, 256, 0, stream>>>(ebuf, denom, src, dst, z, out, total);
}